// GraphFeatureTokenizer_10239202034001
// MI455X (gfx1250) — compile-verified
//
#include <hip/hip_runtime.h>

typedef __attribute__((ext_vector_type(2))) float v2f;
typedef __attribute__((ext_vector_type(8))) float v8f;

#define HID     768
#define KHALF   32          // lap eigvec dim == k
#define KTOT    64          // 2*k
#define MAXLEN  15360
#define NGRAPH  8
#define TILE_T  16
#define TILES_PER_GRAPH (MAXLEN / TILE_T)   // 960
#define COLS_PER_WAVE   (HID / 8)           // 96 (8 waves of 32)
#define NTILES          (COLS_PER_WAVE / 16) // 6 column tiles per wave

#define FEAT_STRIDE 772     // 768 + 4 pad floats: breaks LDS bank aliasing across row halves
#define A_STRIDE    68      // 64 + 4 pad floats: conflict-free A-fragment reads

__global__ __launch_bounds__(256)
void gft_fused_kernel(const float* __restrict__ embw,
                      const float* __restrict__ lapw,
                      const float* __restrict__ ordw,
                      const float* __restrict__ eig,
                      const int*   __restrict__ node_data,
                      const int*   __restrict__ edge_data,
                      const int*   __restrict__ edge_index,
                      const int*   __restrict__ node_num,
                      const int*   __restrict__ edge_num,
                      int E_total,
                      float* __restrict__ out_feat,
                      float* __restrict__ out_mask,
                      float* __restrict__ out_idx)
{
    __shared__ float s_feat[TILE_T * FEAT_STRIDE];   // 49,408 B: embedding sums
    __shared__ float s_A[TILE_T * A_STRIDE];         //  4,352 B: index_embed tile [16][64]
    __shared__ int   s_rows[TILE_T][4];
    __shared__ int   s_src[TILE_T];
    __shared__ int   s_dst[TILE_T];
    __shared__ int   s_ord[TILE_T];
    __shared__ int   s_pad[TILE_T];
    __shared__ int   s_nodebase;

    const int tid = threadIdx.x;
    const int bg  = blockIdx.x / TILES_PER_GRAPH;
    const int t0  = (blockIdx.x % TILES_PER_GRAPH) * TILE_T;

    // ---- Phase 0: per-token metadata (16 threads), mask + index outputs ----
    if (tid < TILE_T) {
        int nb = 0, eb = 0;
        for (int g = 0; g < bg; ++g) { nb += node_num[g]; eb += edge_num[g]; }
        const int nn  = node_num[bg];
        const int seq = nn + edge_num[bg];
        if (tid == 0) s_nodebase = nb;

        const int t = t0 + tid;
        int src = 0, dst = 0, pad = 1;
        int r0 = 0, r1 = 0, r2 = 0, r3 = 0;
        if (t < nn) {                         // node token
            const int gid = nb + t;
            r0 = node_data[gid * 4 + 0]; r1 = node_data[gid * 4 + 1];
            r2 = node_data[gid * 4 + 2]; r3 = node_data[gid * 4 + 3];
            src = t; dst = t; pad = 0;
        } else if (t < seq) {                 // edge token
            const int ge = eb + (t - nn);
            r0 = edge_data[ge * 4 + 0]; r1 = edge_data[ge * 4 + 1];
            r2 = edge_data[ge * 4 + 2]; r3 = edge_data[ge * 4 + 3];
            src = edge_index[ge];             // edge_index[0][e]
            dst = edge_index[E_total + ge];   // edge_index[1][e]
            pad = 0;
        }
        s_rows[tid][0] = r0; s_rows[tid][1] = r1;
        s_rows[tid][2] = r2; s_rows[tid][3] = r3;
        s_src[tid] = src; s_dst[tid] = dst;
        s_ord[tid] = (src == dst) ? 1 : 0;
        s_pad[tid] = pad;

        if (!pad) {   // warm the gathered embedding rows (global_prefetch_b8)
            __builtin_prefetch(embw + (size_t)r0 * HID, 0, 1);
            __builtin_prefetch(embw + (size_t)r1 * HID, 0, 1);
            __builtin_prefetch(embw + (size_t)r2 * HID, 0, 1);
            __builtin_prefetch(embw + (size_t)r3 * HID, 0, 1);
        }

        const long to = (long)bg * MAXLEN + t;
        out_mask[to]        = (t >= seq) ? 1.0f : 0.0f;
        out_idx[to * 2 + 0] = (float)src;
        out_idx[to * 2 + 1] = (float)dst;
    }
    __syncthreads();

    // ---- Phase 1: build index_embed A-tile [16 tokens][64] from eigvecs ----
    {
        const int nb = s_nodebase;
        for (int i = tid; i < TILE_T * KTOT; i += 256) {
            const int m = i >> 6, j = i & 63;
            const int local = (j < KHALF) ? s_src[m] : s_dst[m];
            s_A[m * A_STRIDE + j] = eig[(size_t)(nb + local) * KHALF + (j & 31)];
        }
    }

    // ---- Phase 2: embedding(...).sum(-2) -> s_feat (float4 gathers) ----
    for (int i = tid; i < TILE_T * (HID / 4); i += 256) {
        const int m  = i / (HID / 4);
        const int c4 = i % (HID / 4);
        float4 acc = make_float4(0.f, 0.f, 0.f, 0.f);
        if (!s_pad[m]) {
            const float4* p0 = (const float4*)(embw + (size_t)s_rows[m][0] * HID) + c4;
            const float4* p1 = (const float4*)(embw + (size_t)s_rows[m][1] * HID) + c4;
            const float4* p2 = (const float4*)(embw + (size_t)s_rows[m][2] * HID) + c4;
            const float4* p3 = (const float4*)(embw + (size_t)s_rows[m][3] * HID) + c4;
            float4 a = *p0, b = *p1, c = *p2, d = *p3;
            acc.x = a.x + b.x + c.x + d.x;
            acc.y = a.y + b.y + c.y + d.y;
            acc.z = a.z + b.z + c.z + d.z;
            acc.w = a.w + b.w + c.w + d.w;
        }
        *(float4*)(&s_feat[m * FEAT_STRIDE + c4 * 4]) = acc;
    }
    __syncthreads();

    // ---- Phase 3: [16x64] @ [64x768] via V_WMMA_F32_16X16X4_F32 + epilogue ----
    const int wave = tid >> 5;
    const int lane = tid & 31;
    const int half = lane >> 4;    // 0: lanes 0-15, 1: lanes 16-31
    const int n    = lane & 15;
    const int c0w  = wave * COLS_PER_WAVE;

    // A fragments: lane holds row M=n, K = kk*4 + half*2 + {0,1}
    v2f afr[16];
#pragma unroll
    for (int kk = 0; kk < 16; ++kk) {
        const int kb = kk * 4 + half * 2;
        v2f a;
        a.x = s_A[n * A_STRIDE + kb];
        a.y = s_A[n * A_STRIDE + kb + 1];
        afr[kk] = a;
    }

#pragma unroll
    for (int j = 0; j < NTILES; ++j) {
        const int c0 = c0w + j * 16;
        v8f acc = {};
        const float* bp = lapw + (size_t)(half * 2) * HID + c0 + n;
#pragma unroll
        for (int kk = 0; kk < 16; ++kk) {
            v2f bfr;
            bfr.x = bp[(size_t)(kk * 4) * HID];
            bfr.y = bp[(size_t)(kk * 4 + 1) * HID];
            acc = __builtin_amdgcn_wmma_f32_16x16x4_f32(
                      false, afr[kk], false, bfr, (short)0, acc, false, false);
        }
        // epilogue: + embedding sum + order_weight row, pad-mask, store
#pragma unroll
        for (int v = 0; v < 8; ++v) {
            const int m = v + half * 8;     // C/D layout: VGPR v -> row v (+8 for hi half)
            float val = acc[v]
                      + s_feat[m * FEAT_STRIDE + c0 + n]
                      + ordw[s_ord[m] * HID + c0 + n];
            if (s_pad[m]) val = 0.0f;
            const size_t o = ((size_t)bg * MAXLEN + t0 + m) * (size_t)HID + c0 + n;
            out_feat[o] = val;
        }
    }
}

extern "C" void kernel_launch(void* const* d_in, const int* in_sizes, int n_in,
                              void* d_out, int out_size, void* d_ws, size_t ws_size,
                              hipStream_t stream) {
    const float* embw       = (const float*)d_in[0];
    const float* lapw       = (const float*)d_in[1];
    const float* ordw       = (const float*)d_in[2];
    const float* eig        = (const float*)d_in[3];
    const int*   node_data  = (const int*)d_in[4];
    const int*   edge_data  = (const int*)d_in[5];
    const int*   edge_index = (const int*)d_in[6];
    const int*   node_num   = (const int*)d_in[7];
    const int*   edge_num   = (const int*)d_in[8];
    const int    E_total    = in_sizes[6] / 2;

    float* out_feat = (float*)d_out;
    float* out_mask = out_feat + (size_t)NGRAPH * MAXLEN * HID;
    float* out_idx  = out_mask + (size_t)NGRAPH * MAXLEN;

    dim3 grid(NGRAPH * TILES_PER_GRAPH);   // 7680 workgroups
    dim3 block(256);                       // 8 waves (wave32)
    gft_fused_kernel<<<grid, block, 0, stream>>>(
        embw, lapw, ordw, eig, node_data, edge_data, edge_index,
        node_num, edge_num, E_total, out_feat, out_mask, out_idx);
}